// SetCriterionDynamicKLite_87187836109450
// MI455X (gfx1250) — compile-verified
//
#include <hip/hip_runtime.h>
#include <math.h>

// Problem constants (match the reference).
#define Bdim 32
#define Ndim 2048
#define Cdim 1024
#define Kdim 300
#define NUM_POS (Bdim * Kdim)   // 9600
#define ALPHA_C 0.25f

typedef __attribute__((ext_vector_type(2))) float v2f;
typedef __attribute__((ext_vector_type(8))) float v8f;

// Focal loss for a single logit x against binary target t (0.0f or 1.0f).
// Matches: ce = max(x,0) - x*t + log1p(exp(-|x|)); p = sigmoid(x);
//          p_t = p*t + (1-p)*(1-t); a_t = a*t + (1-a)*(1-t);
//          loss = a_t * ce * (1 - p_t)^2
__device__ __forceinline__ float focal_elem(float x, float t) {
    float ax = fabsf(x);
    float e  = expf(-ax);                       // v_exp_f32 path
    float ce = fmaxf(x, 0.0f) - x * t + log1pf(e);
    // sigmoid via e = exp(-|x|): stable for both signs
    float inv1pe = 1.0f / (1.0f + e);
    float p  = (x >= 0.0f) ? inv1pe : (e * inv1pe);
    float pt = p * t + (1.0f - p) * (1.0f - t);
    float at = ALPHA_C * t + (1.0f - ALPHA_C) * (1.0f - t);
    float om = 1.0f - pt;
    return at * ce * om * om;
}

// One block per (b,k) matched pair. 256 threads; each thread loads a float4
// from the gathered logits row (tid*4 .. tid*4+3) -> entire C=1024 row in one
// coalesced pass. Wave-level reduction of the 64 per-lane partials is done
// with a single V_WMMA_F32_16X16X4_F32 against an all-ones B matrix.
__global__ __launch_bounds__(256)
void focal_gather_kernel(const float* __restrict__ pred_logits,
                         const float* __restrict__ pred_boxes,
                         const float* __restrict__ tgt_boxes,
                         const float* __restrict__ img_sz,
                         const float* __restrict__ img_sz_tgt,
                         const int*   __restrict__ match_idx,
                         const int*   __restrict__ tgt_labels,
                         float* __restrict__ ws_ce,
                         float* __restrict__ ws_bbox,
                         float* __restrict__ ws_giou) {
    const int bk   = blockIdx.x;          // 0 .. NUM_POS-1
    const int b    = bk / Kdim;
    const int q    = match_idx[bk];       // matched query index in [0, N)
    const int lbl  = tgt_labels[bk];      // class in [0, C)
    const int tid  = threadIdx.x;

    const float* row = pred_logits + (size_t)(b * Ndim + q) * Cdim;
    const float4 v = reinterpret_cast<const float4*>(row)[tid];
    const int c0 = tid * 4;

    float l0 = focal_elem(v.x, (c0 + 0) == lbl ? 1.0f : 0.0f);
    float l1 = focal_elem(v.y, (c0 + 1) == lbl ? 1.0f : 0.0f);
    float l2 = focal_elem(v.z, (c0 + 2) == lbl ? 1.0f : 0.0f);
    float l3 = focal_elem(v.w, (c0 + 3) == lbl ? 1.0f : 0.0f);

    // Two f32 partials per lane -> A is the full 16x4 f32 matrix for this wave
    // (64 values). B = ones(4x16) => D[m][n] = sum_k A[m][k] replicated over n.
    float s0 = l0 + l1;
    float s1 = l2 + l3;

    v2f a;    a.x = s0;   a.y = s1;
    v2f ones; ones.x = 1.0f; ones.y = 1.0f;
    v8f cz = {};
#if __has_builtin(__builtin_amdgcn_wmma_f32_16x16x4_f32)
    v8f d = __builtin_amdgcn_wmma_f32_16x16x4_f32(
        /*neg_a=*/false, a, /*neg_b=*/false, ones,
        /*c_mod=*/(short)0, cz, /*reuse_a=*/false, /*reuse_b=*/false);
    // C/D layout: lanes 0-15 VGPR j = D[j][lane]; lanes 16-31 VGPR j = D[j+8][lane-16].
    float hsum = d[0] + d[1] + d[2] + d[3] + d[4] + d[5] + d[6] + d[7];
    float wsum = __shfl(hsum, 0, 32) + __shfl(hsum, 16, 32);
#else
    // Fallback shuffle tree (should not be taken on gfx1250 toolchains).
    float wsum = s0 + s1;
    for (int off = 16; off > 0; off >>= 1)
        wsum += __shfl_down(wsum, off, 32);
    wsum = __shfl(wsum, 0, 32);
#endif

    __shared__ float red[8];
    const int wave = tid >> 5;
    const int lane = tid & 31;
    if (lane == 0) red[wave] = wsum;
    __syncthreads();

    if (tid == 0) {
        float tot = red[0] + red[1] + red[2] + red[3]
                  + red[4] + red[5] + red[6] + red[7];
        ws_ce[bk] = tot;

        // ---- Box losses for this (b,k): tiny, one lane is plenty. ----
        const float* pb = pred_boxes  + (size_t)(b * Ndim + q) * 4;
        const float* gb = tgt_boxes   + (size_t)bk * 4;
        const float* so = img_sz      + (size_t)b * 4;
        const float* st = img_sz_tgt  + (size_t)b * 4;

        float px1 = pb[0], py1 = pb[1], px2 = pb[2], py2 = pb[3];
        float gx1 = gb[0], gy1 = gb[1], gx2 = gb[2], gy2 = gb[3];

        // L1 on size-normalized boxes
        float l1loss = fabsf(px1 / so[0] - gx1 / st[0])
                     + fabsf(py1 / so[1] - gy1 / st[1])
                     + fabsf(px2 / so[2] - gx2 / st[2])
                     + fabsf(py2 / so[3] - gy2 / st[3]);
        ws_bbox[bk] = l1loss;

        // GIoU in pixel space
        float area1 = (px2 - px1) * (py2 - py1);
        float area2 = (gx2 - gx1) * (gy2 - gy1);
        float ltx = fmaxf(px1, gx1), lty = fmaxf(py1, gy1);
        float rbx = fminf(px2, gx2), rby = fminf(py2, gy2);
        float iw  = fmaxf(rbx - ltx, 0.0f);
        float ih  = fmaxf(rby - lty, 0.0f);
        float inter = iw * ih;
        float uni   = area1 + area2 - inter;
        float iou   = inter / uni;
        float cx1 = fminf(px1, gx1), cy1 = fminf(py1, gy1);
        float cx2 = fmaxf(px2, gx2), cy2 = fmaxf(py2, gy2);
        float cw  = fmaxf(cx2 - cx1, 0.0f);
        float ch  = fmaxf(cy2 - cy1, 0.0f);
        float areac = cw * ch;
        float giou  = iou - (areac - uni) / areac;
        ws_giou[bk] = 1.0f - giou;
    }
}

// Deterministic single-block final reduction of the 9600 per-pair partials.
__global__ __launch_bounds__(256)
void finalize_kernel(const float* __restrict__ ws_ce,
                     const float* __restrict__ ws_bbox,
                     const float* __restrict__ ws_giou,
                     float* __restrict__ out) {
    __shared__ float sce[256], sbb[256], sgi[256];
    const int tid = threadIdx.x;
    float ace = 0.0f, abb = 0.0f, agi = 0.0f;
    for (int i = tid; i < NUM_POS; i += 256) {
        ace += ws_ce[i];
        abb += ws_bbox[i];
        agi += ws_giou[i];
    }
    sce[tid] = ace; sbb[tid] = abb; sgi[tid] = agi;
    __syncthreads();
    for (int s = 128; s > 0; s >>= 1) {
        if (tid < s) {
            sce[tid] += sce[tid + s];
            sbb[tid] += sbb[tid + s];
            sgi[tid] += sgi[tid + s];
        }
        __syncthreads();
    }
    if (tid == 0) {
        const float inv = 1.0f / (float)NUM_POS;
        out[0] = sce[0] * inv;                      // loss_ce
        out[1] = sbb[0] * inv;                      // loss_bbox
        out[2] = sgi[0] * inv;                      // loss_giou
        out[3] = (float)Kdim / (float)Ndim;         // matched_frac
    }
}

extern "C" void kernel_launch(void* const* d_in, const int* in_sizes, int n_in,
                              void* d_out, int out_size, void* d_ws, size_t ws_size,
                              hipStream_t stream) {
    const float* pred_logits = (const float*)d_in[0];
    const float* pred_boxes  = (const float*)d_in[1];
    const float* tgt_boxes   = (const float*)d_in[2];
    const float* img_sz      = (const float*)d_in[3];
    const float* img_sz_tgt  = (const float*)d_in[4];
    const int*   match_idx   = (const int*)d_in[5];
    const int*   tgt_labels  = (const int*)d_in[6];

    float* ws      = (float*)d_ws;
    float* ws_ce   = ws;
    float* ws_bbox = ws + NUM_POS;
    float* ws_giou = ws + 2 * NUM_POS;

    focal_gather_kernel<<<NUM_POS, 256, 0, stream>>>(
        pred_logits, pred_boxes, tgt_boxes, img_sz, img_sz_tgt,
        match_idx, tgt_labels, ws_ce, ws_bbox, ws_giou);

    finalize_kernel<<<1, 256, 0, stream>>>(ws_ce, ws_bbox, ws_giou, (float*)d_out);
}